// LaneAwareCrossEntropyLoss_59863254172422
// MI455X (gfx1250) — compile-verified
//
#include <hip/hip_runtime.h>

// Reference: B=65536, C=2, P=41, L=4; loss = mean(ce*boundary_w + 0.001*connectivity)
// Bandwidth-bound single-pass reduction (~129 MB @ 23.3 TB/s ~ 5.5us floor).

#define ALPHA 1.0f      // GAMMA
#define BETA  0.5f      // GAMMA * 0.5
#define CW    0.001f    // CONNECTIVITY_WEIGHT

typedef __attribute__((ext_vector_type(2))) float v2f;
typedef __attribute__((ext_vector_type(8))) float v8f;

__launch_bounds__(256)
__global__ void ce_boundary_loss_kernel(const float* __restrict__ logits,
                                        const int*   __restrict__ targets,
                                        float* __restrict__ out,
                                        int B, float invN) {
    constexpr int C = 2, P = 41, L = 4;

    const int t     = blockIdx.x * blockDim.x + threadIdx.x;
    const bool valid = t < B * L;
    const int tt    = valid ? t : 0;          // invalid lanes redo lane-0 work (EXEC stays full for WMMA)
    const int b     = tt >> 2;
    const int l     = tt & 3;

    // thread (b,l): logits[b, c, p, l] = base + c*P*L + p*L ; consecutive lanes -> consecutive dwords
    const float* lp = logits  + (size_t)b * (C * P * L) + l;
    const int*   tp = targets + (size_t)b * (P * L) + l;

    float acc = 0.0f;
    float loss_m1 = 0.0f, loss_m2 = 0.0f;
    float lab_m1  = 0.0f, lab_m2  = 0.0f;
    int   pred_m1 = 0;

    for (int p = 0; p < P; ++p) {
        const float x0 = __builtin_nontemporal_load(lp + p * L);
        const float x1 = __builtin_nontemporal_load(lp + P * L + p * L);
        const int   tg = __builtin_nontemporal_load(tp + p * L);

        const float d    = fabsf(x0 - x1);
        const int   pred = (x1 > x0) ? 1 : 0;              // argmax (first-max ties -> 0)
        const float gap  = (tg == pred) ? 0.0f : d;        // max - x_target
        const float loss = gap + log1pf(__expf(-d));       // lse - x_target
        const float lab  = (float)tg;

        // w[p] stencil expanded: sum loss + alpha*d1*(l_i+l_{i+1}) + beta*d2*(l_i+l_{i+2})
        acc += loss;
        if (p >= 1) {
            const float d1 = fabsf(lab - lab_m1);
            acc += ALPHA * d1 * (loss + loss_m1);
            acc += (pred != pred_m1) ? CW : 0.0f;          // connectivity
        }
        if (p >= 2) {
            const float d2 = fabsf(lab - lab_m2);
            acc += BETA * d2 * (loss + loss_m2);
        }
        loss_m2 = loss_m1; loss_m1 = loss;
        lab_m2  = lab_m1;  lab_m1  = lab;
        pred_m1 = pred;
    }
    if (!valid) acc = 0.0f;

    // ---- wave32 reduction via V_WMMA_F32_16X16X4_F32 ----
    // A = ones(16x4), B: b[0]=partial, b[1]=0  => D[m][n] = sum_k B[k][n] = s_n + s_{n+16}
    // (independent of B's row-to-VGPR mapping; only column = lane%16 matters)
    v2f amat; amat[0] = 1.0f; amat[1] = 1.0f;
    v2f bmat; bmat[0] = acc;  bmat[1] = 0.0f;
    v8f cz = {};
    v8f dres = __builtin_amdgcn_wmma_f32_16x16x4_f32(
        /*neg_a=*/false, amat, /*neg_b=*/false, bmat,
        /*c_mod=*/(short)0, cz, /*reuse_a=*/false, /*reuse_b=*/false);
    float x = dres[0];                        // every lane: s_{lane%16} + s_{lane%16+16}
    x += __shfl_xor(x, 8, 32);
    x += __shfl_xor(x, 4, 32);
    x += __shfl_xor(x, 2, 32);
    x += __shfl_xor(x, 1, 32);                // full wave sum on all lanes

    __shared__ float wsum[8];
    const int lane = threadIdx.x & 31;
    const int wid  = threadIdx.x >> 5;
    if (lane == 0) wsum[wid] = x;
    __syncthreads();
    if (threadIdx.x == 0) {
        float s = 0.0f;
        #pragma unroll
        for (int i = 0; i < 8; ++i) s += wsum[i];
        atomicAdd(out, s * invN);
    }
}

extern "C" void kernel_launch(void* const* d_in, const int* in_sizes, int n_in,
                              void* d_out, int out_size, void* d_ws, size_t ws_size,
                              hipStream_t stream) {
    const float* logits  = (const float*)d_in[0];
    const int*   targets = (const int*)d_in[1];   // jnp "int64" -> int32 (x64 disabled), per harness
    float* out = (float*)d_out;

    const int B = in_sizes[0] / (2 * 41 * 4);
    const int total = B * 4;
    const int threads = 256;
    const int blocks = (total + threads - 1) / threads;
    const float invN = 1.0f / (float)(B * 41 * 4);

    hipMemsetAsync(out, 0, sizeof(float), stream);
    ce_boundary_loss_kernel<<<blocks, threads, 0, stream>>>(logits, targets, out, B, invN);
}